// GraphEncoder_75565654606173
// MI455X (gfx1250) — compile-verified
//
#include <hip/hip_runtime.h>
#include <math.h>

// ---------------- problem constants ----------------
#define NND   100000      // nodes
#define NUND  1600000     // undirected edges
#define EDIR  3200000     // directed edges (2x undirected)
#define ETOT  3300000     // directed + self loops  (multiple of 16)
#define NGR   512         // graphs
#define LAT1  24
#define LAT2  36
#define LATE  12
#define ODIM  64

typedef __attribute__((ext_vector_type(2))) float v2f;
typedef __attribute__((ext_vector_type(8))) float v8f;

// ---------------- WMMA wrapper: fp32 16x16x4 ----------------
__device__ __forceinline__ v8f wmma4(v2f a, v2f b, v8f c) {
  return __builtin_amdgcn_wmma_f32_16x16x4_f32(false, a, false, b, (short)0, c,
                                               false, false);
}

// Generic tiled linear: out[rows,N] = in[rows,K] @ W[N,K]^T (+ bias)
// rows multiple of 16; K multiple of 4 (and even -> 8B-aligned v2f loads);
// N arbitrary: pad columns are CLAMPED to col N-1 (their D columns are never
// stored, and WMMA columns are independent, so no zero-masking is needed).
// A frag (16x4 fp32): lane l -> M = l&15, K = kc + 2*(l>>4) + {0,1}
// B frag (4x16 fp32): lane l -> N = l&15, K = kc + 2*(l>>4) + {0,1}
// C/D frag          : lane l, vgpr v -> M = v + 8*(l>>4), N = l&15
template <int K, int N>
__global__ __launch_bounds__(256) void linear_wmma(
    const float* __restrict__ in, const float* __restrict__ W,
    const float* __restrict__ bias, float* __restrict__ out, int rows) {
  constexpr int colTiles = (N + 15) / 16;
  const int lane = threadIdx.x & 31;
  const int m    = lane & 15;
  const int hi   = lane >> 4;
  const int wave = (blockIdx.x * blockDim.x + threadIdx.x) >> 5;
  const int nw   = (gridDim.x * blockDim.x) >> 5;
  const int rowTiles = rows >> 4;
  const int total    = rowTiles * colTiles;

  for (int t = wave; t < total; t += nw) {
    const int rt   = t / colTiles;
    const int ct   = t - rt * colTiles;
    const int col  = ct * 16 + m;
    const bool cok = (col < N);
    const int colc = cok ? col : (N - 1);   // clamp pad columns (never stored)

    const float bv = (bias != nullptr) ? bias[colc] : 0.0f;
    v8f c;
#pragma unroll
    for (int v = 0; v < 8; ++v) c[v] = bv;

    const float* arow = in + (size_t)(rt * 16 + m) * K + 2 * hi;
    const float* wrow = W + (size_t)colc * K + 2 * hi;

#pragma unroll
    for (int kc = 0; kc < K; kc += 4) {
      v2f a = *(const v2f*)(arow + kc);   // global_load_b64
      v2f b = *(const v2f*)(wrow + kc);   // global_load_b64
      c = wmma4(a, b, c);
    }

    float* orow = out + (size_t)(rt * 16 + 8 * hi) * N + col;
#pragma unroll
    for (int v = 0; v < 8; ++v) {
      if (cok) orow[(size_t)v * N] = c[v];
    }
  }
}

// ---------------- helpers ----------------
__device__ __forceinline__ void edge_sd(int e, const int* __restrict__ ei,
                                        int& s, int& d) {
  if (e < EDIR) { s = ei[e]; d = ei[EDIR + e]; }
  else          { s = e - EDIR; d = s; }   // appended self loops
}

__device__ __forceinline__ void atomicMaxF(float* addr, float val) {
  int* ai = (int*)addr;
  int old = __float_as_int(*addr);
  while (__int_as_float(old) < val) {
    int assumed = old;
    old = atomicCAS(ai, assumed, __float_as_int(val));
    if (old == assumed) break;
  }
}

__global__ void fill_f32(float* __restrict__ p, size_t n, float v) {
  size_t i = (size_t)blockIdx.x * blockDim.x + threadIdx.x;
  if (i < n) p[i] = v;
}

// ---------------- node embedding + 17->24 linear ----------------
__global__ void node_embed_kernel(const float* __restrict__ x,
                                  const float* __restrict__ emb,
                                  const float* __restrict__ nw,
                                  const float* __restrict__ nb,
                                  float* __restrict__ h) {
  int i = blockIdx.x * blockDim.x + threadIdx.x;
  if (i >= NND) return;
  const float* xr = x + (size_t)i * 119;
  int best = 0; float bv = xr[0];
#pragma unroll 4
  for (int j = 1; j < 118; ++j) { float v = xr[j]; if (v > bv) { bv = v; best = j; } }
  float feat[17];
#pragma unroll
  for (int k = 0; k < 16; ++k) feat[k] = emb[best * 16 + k];
  feat[16] = xr[118];
  float* ho = h + (size_t)i * LAT1;
#pragma unroll
  for (int o = 0; o < LAT1; ++o) {
    const float* wr = nw + o * 17;
    float acc = nb[o];
#pragma unroll
    for (int k = 0; k < 17; ++k) acc = fmaf(wr[k], feat[k], acc);
    ho[o] = acc;
  }
}

// ------- edge latent (4->12) + RAW attr mean accumulation (linearity) -------
__global__ void edge_latent_kernel(const float* __restrict__ eattr,
                                   const int* __restrict__ ei,
                                   const float* __restrict__ ew,
                                   const float* __restrict__ eb,
                                   float* __restrict__ eaL,
                                   float* __restrict__ ssum4,
                                   float* __restrict__ cnt) {
  int e = blockIdx.x * blockDim.x + threadIdx.x;
  if (e >= EDIR) return;
  int ae = (e >= NUND) ? e - NUND : e;   // duplicated attrs for reverse edges
  const float* a = eattr + (size_t)ae * 4;
  const float a0 = a[0], a1 = a[1], a2 = a[2], a3 = a[3];
  const int dst = ei[EDIR + e];
  float* outp = eaL + (size_t)e * LATE;
#pragma unroll
  for (int j = 0; j < LATE; ++j) {
    const float* w = ew + j * 4;
    outp[j] = eb[j] + w[0] * a0 + w[1] * a1 + w[2] * a2 + w[3] * a3;
  }
  float* sp = ssum4 + (size_t)dst * 4;
  atomicAdd(sp + 0, a0);
  atomicAdd(sp + 1, a1);
  atomicAdd(sp + 2, a2);
  atomicAdd(sp + 3, a3);
  atomicAdd(cnt + dst, 1.0f);
}

// self-loop attr = mean(incoming latent) = edge_w @ mean(raw) + edge_b
// (exactly, by linearity). Isolated nodes (cnt==0) get the zero vector,
// matching the reference's segment_sum/max(cnt,1) semantics.
__global__ void self_loop_kernel(const float* __restrict__ ssum4,
                                 const float* __restrict__ cnt,
                                 const float* __restrict__ ew,
                                 const float* __restrict__ eb,
                                 float* __restrict__ eaL_self) {
  int i = blockIdx.x * blockDim.x + threadIdx.x;
  if (i >= NND) return;
  float c = cnt[i];
  float* outp = eaL_self + (size_t)i * LATE;
  if (c > 0.0f) {
    float inv = 1.0f / c;
    float m0 = ssum4[(size_t)i * 4 + 0] * inv;
    float m1 = ssum4[(size_t)i * 4 + 1] * inv;
    float m2 = ssum4[(size_t)i * 4 + 2] * inv;
    float m3 = ssum4[(size_t)i * 4 + 3] * inv;
#pragma unroll
    for (int j = 0; j < LATE; ++j) {
      const float* w = ew + j * 4;
      outp[j] = eb[j] + w[0] * m0 + w[1] * m1 + w[2] * m2 + w[3] * m3;
    }
  } else {
#pragma unroll
    for (int j = 0; j < LATE; ++j) outp[j] = 0.0f;
  }
}

// ---------------- attention passes ----------------
template <int D>
__global__ void edge_logits_kernel(const float* __restrict__ xl,
                                   const float* __restrict__ xr,
                                   const float* __restrict__ ew,
                                   const float* __restrict__ att,
                                   const int* __restrict__ ei,
                                   float* __restrict__ logit,
                                   float* __restrict__ maxd) {
  int e = blockIdx.x * blockDim.x + threadIdx.x;
  if (e >= ETOT) return;
  int s, d; edge_sd(e, ei, s, d);
  const float* pl = xl + (size_t)s * D;
  const float* pr = xr + (size_t)d * D;
  const float* pe = ew + (size_t)e * D;
  float acc = 0.0f;
#pragma unroll
  for (int j = 0; j < D; ++j) {
    float m  = pl[j] + pr[j] + pe[j];
    float lr = (m > 0.0f) ? m : 0.2f * m;   // leaky_relu, slope 0.2
    acc = fmaf(lr, att[j], acc);
  }
  logit[e] = acc;
  atomicMaxF(maxd + d, acc);
}

__global__ void edge_expsum_kernel(float* __restrict__ logit,
                                   const float* __restrict__ maxd,
                                   float* __restrict__ sumd,
                                   const int* __restrict__ ei) {
  int e = blockIdx.x * blockDim.x + threadIdx.x;
  if (e >= ETOT) return;
  int s, d; edge_sd(e, ei, s, d);
  float v = expf(logit[e] - maxd[d]);
  logit[e] = v;                 // reuse buffer for exp values
  atomicAdd(sumd + d, v);
}

template <int D>
__global__ void edge_agg_kernel(const float* __restrict__ el,
                                const float* __restrict__ sumd,
                                const float* __restrict__ xl,
                                const int* __restrict__ ei,
                                float* __restrict__ agg) {
  int e = blockIdx.x * blockDim.x + threadIdx.x;
  if (e >= ETOT) return;
  int s, d; edge_sd(e, ei, s, d);
  float alpha = el[e] / sumd[d];
  const float* pl = xl + (size_t)s * D;
  float* pa = agg + (size_t)d * D;
#pragma unroll
  for (int j = 0; j < D; ++j) atomicAdd(pa + j, alpha * pl[j]);
}

template <int D>
__global__ void node_bias_relu_kernel(const float* __restrict__ agg,
                                      const float* __restrict__ bias,
                                      float* __restrict__ h) {
  size_t i = (size_t)blockIdx.x * blockDim.x + threadIdx.x;
  if (i >= (size_t)NND * D) return;
  int j = (int)(i % D);
  float v = agg[i] + bias[j];
  h[i] = v > 0.0f ? v : 0.0f;
}

// ---------------- pooling ----------------
__global__ void pool_sum_kernel(const float* __restrict__ h2,
                                const int* __restrict__ batch,
                                float* __restrict__ pooled,
                                float* __restrict__ gcnt) {
  int i = blockIdx.x * blockDim.x + threadIdx.x;
  if (i >= NND) return;
  int g = batch[i];
  const float* hr = h2 + (size_t)i * LAT2;
  float* pr = pooled + (size_t)g * LAT2;
#pragma unroll
  for (int j = 0; j < LAT2; ++j) atomicAdd(pr + j, hr[j]);
  atomicAdd(gcnt + g, 1.0f);
}

__global__ void pool_div_kernel(float* __restrict__ pooled,
                                const float* __restrict__ gcnt) {
  int i = blockIdx.x * blockDim.x + threadIdx.x;
  if (i >= NGR * LAT2) return;
  pooled[i] /= fmaxf(gcnt[i / LAT2], 1.0f);
}

// ---------------- host driver ----------------
extern "C" void kernel_launch(void* const* d_in, const int* in_sizes, int n_in,
                              void* d_out, int out_size, void* d_ws,
                              size_t ws_size, hipStream_t stream) {
  const float* x       = (const float*)d_in[0];
  const int*   ei      = (const int*)d_in[1];
  const float* eattr   = (const float*)d_in[2];
  const int*   batch   = (const int*)d_in[3];
  const float* emb     = (const float*)d_in[4];
  const float* node_w  = (const float*)d_in[5];
  const float* node_b  = (const float*)d_in[6];
  const float* edge_w  = (const float*)d_in[7];
  const float* edge_b  = (const float*)d_in[8];
  const float* c1_wl   = (const float*)d_in[9];
  const float* c1_bl   = (const float*)d_in[10];
  const float* c1_wr   = (const float*)d_in[11];
  const float* c1_br   = (const float*)d_in[12];
  const float* c1_we   = (const float*)d_in[13];
  const float* c1_att  = (const float*)d_in[14];
  const float* c1_bias = (const float*)d_in[15];
  const float* c2_wl   = (const float*)d_in[16];
  const float* c2_bl   = (const float*)d_in[17];
  const float* c2_wr   = (const float*)d_in[18];
  const float* c2_br   = (const float*)d_in[19];
  const float* c2_we   = (const float*)d_in[20];
  const float* c2_att  = (const float*)d_in[21];
  const float* c2_bias = (const float*)d_in[22];
  const float* final_w = (const float*)d_in[23];
  const float* final_b = (const float*)d_in[24];
  float* out = (float*)d_out;

  // -------- workspace carve (bump allocator, 256B aligned) --------
  char* ws = (char*)d_ws;
  size_t off = 0;
  auto alloc = [&](size_t nfloats) -> float* {
    float* p = (float*)(ws + off);
    off += ((nfloats * sizeof(float) + 255) / 256) * 256;
    return p;
  };
  float* h       = alloc((size_t)NND * LAT1);
  float* eaL     = alloc((size_t)ETOT * LATE);
  float* ssum4   = alloc((size_t)NND * 4);
  float* cnt     = alloc(NND);
  float* ew1     = alloc((size_t)ETOT * LAT1);
  float* ew2     = alloc((size_t)ETOT * LAT2);
  float* xl1     = alloc((size_t)NND * LAT1);
  float* xr1     = alloc((size_t)NND * LAT1);
  float* logit   = alloc(ETOT);
  float* maxd    = alloc(NND);
  float* sumd    = alloc(NND);
  float* agg1    = alloc((size_t)NND * LAT1);
  float* h1      = alloc((size_t)NND * LAT1);
  float* xl2     = alloc((size_t)NND * LAT2);
  float* xr2     = alloc((size_t)NND * LAT2);
  float* agg2    = alloc((size_t)NND * LAT2);
  float* h2      = alloc((size_t)NND * LAT2);
  float* pooled  = alloc((size_t)NGR * LAT2);
  float* gcnt    = alloc(NGR);
  (void)ws_size; (void)in_sizes; (void)n_in; (void)out_size;

  auto fill = [&](float* p, size_t n, float v) {
    int blocks = (int)((n + 255) / 256);
    fill_f32<<<blocks, 256, 0, stream>>>(p, n, v);
  };
  auto lin_grid = [](int rows, int N) {
    long waves  = (long)(rows >> 4) * ((N + 15) >> 4);
    long blocks = (waves + 7) / 8;          // 8 waves / 256-thread block
    if (blocks < 1) blocks = 1;
    if (blocks > 131072) blocks = 131072;   // grid-stride covers the rest
    return (int)blocks;
  };
  const int EB  = (EDIR + 255) / 256;
  const int ETB = (ETOT + 255) / 256;
  const int NB  = (NND + 255) / 256;

  // ---- node features ----
  node_embed_kernel<<<NB, 256, 0, stream>>>(x, emb, node_w, node_b, h);

  // ---- edge latent features + self-loop mean ----
  fill(ssum4, (size_t)NND * 4, 0.0f);
  fill(cnt, NND, 0.0f);
  edge_latent_kernel<<<EB, 256, 0, stream>>>(eattr, ei, edge_w, edge_b, eaL,
                                             ssum4, cnt);
  self_loop_kernel<<<NB, 256, 0, stream>>>(ssum4, cnt, edge_w, edge_b,
                                           eaL + (size_t)EDIR * LATE);

  // ---- batched WMMA GEMMs: edge-weight projections (both layers) ----
  linear_wmma<LATE, LAT1><<<lin_grid(ETOT, LAT1), 256, 0, stream>>>(
      eaL, c1_we, nullptr, ew1, ETOT);
  linear_wmma<LATE, LAT2><<<lin_grid(ETOT, LAT2), 256, 0, stream>>>(
      eaL, c2_we, nullptr, ew2, ETOT);

  // ================= GATv2 layer 1 (24 -> 24) =================
  linear_wmma<LAT1, LAT1><<<lin_grid(NND, LAT1), 256, 0, stream>>>(
      h, c1_wl, c1_bl, xl1, NND);
  linear_wmma<LAT1, LAT1><<<lin_grid(NND, LAT1), 256, 0, stream>>>(
      h, c1_wr, c1_br, xr1, NND);
  fill(maxd, NND, -3.402823466e38f);
  fill(sumd, NND, 0.0f);
  fill(agg1, (size_t)NND * LAT1, 0.0f);
  edge_logits_kernel<LAT1><<<ETB, 256, 0, stream>>>(xl1, xr1, ew1, c1_att, ei,
                                                    logit, maxd);
  edge_expsum_kernel<<<ETB, 256, 0, stream>>>(logit, maxd, sumd, ei);
  edge_agg_kernel<LAT1><<<ETB, 256, 0, stream>>>(logit, sumd, xl1, ei, agg1);
  node_bias_relu_kernel<LAT1>
      <<<(int)(((size_t)NND * LAT1 + 255) / 256), 256, 0, stream>>>(agg1,
                                                                    c1_bias, h1);

  // ================= GATv2 layer 2 (24 -> 36) =================
  linear_wmma<LAT1, LAT2><<<lin_grid(NND, LAT2), 256, 0, stream>>>(
      h1, c2_wl, c2_bl, xl2, NND);
  linear_wmma<LAT1, LAT2><<<lin_grid(NND, LAT2), 256, 0, stream>>>(
      h1, c2_wr, c2_br, xr2, NND);
  fill(maxd, NND, -3.402823466e38f);
  fill(sumd, NND, 0.0f);
  fill(agg2, (size_t)NND * LAT2, 0.0f);
  edge_logits_kernel<LAT2><<<ETB, 256, 0, stream>>>(xl2, xr2, ew2, c2_att, ei,
                                                    logit, maxd);
  edge_expsum_kernel<<<ETB, 256, 0, stream>>>(logit, maxd, sumd, ei);
  edge_agg_kernel<LAT2><<<ETB, 256, 0, stream>>>(logit, sumd, xl2, ei, agg2);
  node_bias_relu_kernel<LAT2>
      <<<(int)(((size_t)NND * LAT2 + 255) / 256), 256, 0, stream>>>(agg2,
                                                                    c2_bias, h2);

  // ---- mean pool + final linear (WMMA, writes d_out) ----
  fill(pooled, (size_t)NGR * LAT2, 0.0f);
  fill(gcnt, NGR, 0.0f);
  pool_sum_kernel<<<NB, 256, 0, stream>>>(h2, batch, pooled, gcnt);
  pool_div_kernel<<<(NGR * LAT2 + 255) / 256, 256, 0, stream>>>(pooled, gcnt);
  linear_wmma<LAT2, ODIM><<<lin_grid(NGR, ODIM), 256, 0, stream>>>(
      pooled, final_w, final_b, out, NGR);
}